// Attention_39015482916872
// MI455X (gfx1250) — compile-verified
//
#include <hip/hip_runtime.h>
#include <math.h>

// ---------------------------------------------------------------------------
// CDNA5 (gfx1250) fp32 attention via V_WMMA_F32_16X16X4_F32 (wave32).
// Shapes: B=2, N=2048, C=1024, H=16, D=64.
// ---------------------------------------------------------------------------

typedef float v2f __attribute__((ext_vector_type(2)));
typedef float v8f __attribute__((ext_vector_type(8)));

// D(16x16,f32) = A(16x4,f32) * B(4x16,f32) + C
// A layout per lane: a.x = A[lane&15][2*(lane>>4)], a.y = A[lane&15][2*(lane>>4)+1]
// B layout per lane: b.x = B[2*(lane>>4)][lane&15], b.y = B[2*(lane>>4)+1][lane&15]
// C/D layout: slot r, lanes 0-15 -> row r, lanes 16-31 -> row r+8; col = lane&15
__device__ __forceinline__ v8f wmma_f32x4(v2f a, v2f b, v8f c) {
  return __builtin_amdgcn_wmma_f32_16x16x4_f32(
      /*neg_a=*/false, a, /*neg_b=*/false, b,
      /*c_mod=*/(short)0, c, /*reuse_a=*/false, /*reuse_b=*/false);
}

// ---------------------------------------------------------------------------
// LDS-staged GEMM: Cmat[m][n] = sum_k A[m][k] * Bw[n][k] (+ bias[n])
// A: (M,K) row-major.  Bw: (N,K) row-major (computes A @ Bw^T).
// Block = 256 threads = 8 waves; block tile 128(M) x 64(N); K-chunk 32.
// A/B panels double-buffered in LDS; global loads for chunk kc+1 issue before
// the 32 WMMAs of chunk kc, so vmem latency amortizes over a whole panel.
// LDS pitch 36 dwords -> ds_load_b64 fragment reads hit all 64 banks once.
// Grid: (Nn/64, M/128).
// ---------------------------------------------------------------------------
constexpr int KT = 32;  // K-chunk
constexpr int LP = 36;  // LDS pitch (pad vs 32)

__global__ __launch_bounds__(256) void gemm16_wmma_f32(
    const float* __restrict__ A, const float* __restrict__ Bw,
    const float* __restrict__ bias, float* __restrict__ Cmat,
    int M, int Nn, int K) {
  __shared__ float asmem[2][128 * LP];  // 2 x 18 KB
  __shared__ float bsmem[2][64 * LP];   // 2 x  9 KB

  const int tid   = threadIdx.x;
  const int lane  = tid & 31;
  const int wave  = tid >> 5;
  const int laneM = lane & 15;
  const int laneH = lane >> 4;

  const int rowblk = blockIdx.y * 128;
  const int col0   = blockIdx.x * 64;

  // cooperative staging maps (per thread)
  const int arow_t = tid >> 1;        // 0..127 : A row
  const int ahalf  = (tid & 1) * 16;  // 0/16   : A k-offset (16 floats)
  const int bcol_t = tid >> 2;        // 0..63  : B row (output col)
  const int bq     = (tid & 3) * 8;   // 0..24  : B k-offset (8 floats)

  const float* ag = A  + (size_t)(rowblk + arow_t) * K + ahalf;
  const float* bg = Bw + (size_t)(col0   + bcol_t) * K + bq;

  float4 a_r[4], b_r[2];
  #pragma unroll
  for (int i = 0; i < 4; ++i) a_r[i] = *(const float4*)(ag + 4 * i);
  #pragma unroll
  for (int i = 0; i < 2; ++i) b_r[i] = *(const float4*)(bg + 4 * i);
  #pragma unroll
  for (int i = 0; i < 4; ++i)
    *(float4*)(&asmem[0][arow_t * LP + ahalf + 4 * i]) = a_r[i];
  #pragma unroll
  for (int i = 0; i < 2; ++i)
    *(float4*)(&bsmem[0][bcol_t * LP + bq + 4 * i]) = b_r[i];
  __syncthreads();

  v8f acc0 = {0.f, 0.f, 0.f, 0.f, 0.f, 0.f, 0.f, 0.f};
  v8f acc1 = acc0, acc2 = acc0, acc3 = acc0;

  const int NC = K / KT;
  for (int kc = 0; kc < NC; ++kc) {
    const int cur = kc & 1;

    // issue global loads for the NEXT panel before compute
    if (kc + 1 < NC) {
      const float* agn = ag + (size_t)(kc + 1) * KT;
      const float* bgn = bg + (size_t)(kc + 1) * KT;
      #pragma unroll
      for (int i = 0; i < 4; ++i) a_r[i] = *(const float4*)(agn + 4 * i);
      #pragma unroll
      for (int i = 0; i < 2; ++i) b_r[i] = *(const float4*)(bgn + 4 * i);
    }

    // ---- 8 k-steps x 4 WMMAs from LDS ------------------------------------
    const float* as  = &asmem[cur][(wave * 16 + laneM) * LP + 2 * laneH];
    const float* bs0 = &bsmem[cur][laneM * LP + 2 * laneH];
    const float* bs1 = bs0 + 16 * LP;
    const float* bs2 = bs0 + 32 * LP;
    const float* bs3 = bs0 + 48 * LP;
    #pragma unroll
    for (int k = 0; k < KT; k += 4) {
      v2f af  = *(const v2f*)(as  + k);
      v2f bf0 = *(const v2f*)(bs0 + k);
      v2f bf1 = *(const v2f*)(bs1 + k);
      v2f bf2 = *(const v2f*)(bs2 + k);
      v2f bf3 = *(const v2f*)(bs3 + k);
      acc0 = wmma_f32x4(af, bf0, acc0);
      acc1 = wmma_f32x4(af, bf1, acc1);
      acc2 = wmma_f32x4(af, bf2, acc2);
      acc3 = wmma_f32x4(af, bf3, acc3);
    }

    // ---- store staged panel, flip buffers --------------------------------
    if (kc + 1 < NC) {
      const int nxt = cur ^ 1;
      #pragma unroll
      for (int i = 0; i < 4; ++i)
        *(float4*)(&asmem[nxt][arow_t * LP + ahalf + 4 * i]) = a_r[i];
      #pragma unroll
      for (int i = 0; i < 2; ++i)
        *(float4*)(&bsmem[nxt][bcol_t * LP + bq + 4 * i]) = b_r[i];
    }
    __syncthreads();
  }

  float bv0 = 0.f, bv1 = 0.f, bv2 = 0.f, bv3 = 0.f;
  if (bias) {
    bv0 = bias[col0 + laneM];
    bv1 = bias[col0 + 16 + laneM];
    bv2 = bias[col0 + 32 + laneM];
    bv3 = bias[col0 + 48 + laneM];
  }
  const int row0 = rowblk + wave * 16;
  #pragma unroll
  for (int r = 0; r < 8; ++r) {
    const int row = row0 + r + 8 * laneH;
    float* crow = Cmat + (size_t)row * Nn + col0 + laneM;
    crow[0]  = acc0[r] + bv0;
    crow[16] = acc1[r] + bv1;
    crow[32] = acc2[r] + bv2;
    crow[48] = acc3[r] + bv3;
  }
}

// ---------------------------------------------------------------------------
// Flash attention. All 8 waves of a block share (b,h) and sweep the same key
// blocks, so K (RoPE'd) and V tiles are staged cooperatively in LDS, double
// buffered: global loads for block j+1 issue before the 32 WMMAs of block j.
// K tile: row-major, pitch 68 (conflict-free ds_load_b64 fragments).
// V tile: pair-interleaved [k/2][n][2], pitch 130 -> each B fragment is one
// aligned conflict-free ds_load_b64.
// qkv: (B, N, 3*C): cols [0,1024)=Q, [1024,2048)=K, [2048,3072)=V (h*64+d).
// freqs: (N, 32, 2) = (cos, sin). out: (B, N, C).
// Grid: (N/16/8, H, B); block = 256 = 8 waves; 1 wave per 16-query block.
// ---------------------------------------------------------------------------
constexpr int NTOK = 2048;
constexpr int QKVS = 3072;
constexpr int CCH  = 1024;
constexpr int KVP  = 68;   // K tile pitch
constexpr int VPP  = 130;  // V pair-row pitch (pad vs 128)
constexpr int PP   = 18;   // P tile pitch (even -> aligned b64)

struct KVregs {
  v2f t0, t1, c0, c1;  // raw K pairs + (cos,sin) pairs
  float4 vv;           // raw V 4-vector
};

__device__ __forceinline__ KVregs stage_load(
    const float* __restrict__ qkv, const float* __restrict__ freqs,
    int b, int h, int j, int tid) {
  KVregs r;
  const int tok16  = tid >> 4;   // 0..15: token within tile
  const int lane16 = tid & 15;   // 0..15: rope-pair / d-group index
  const int token  = j * 16 + tok16;
  const float* kg = qkv + ((size_t)(b * NTOK + token)) * QKVS + CCH + h * 64;
  r.t0 = *(const v2f*)(kg + 2 * lane16);
  r.t1 = *(const v2f*)(kg + 2 * lane16 + 32);
  r.c0 = *(const v2f*)(freqs + ((size_t)token * 32 + lane16) * 2);
  r.c1 = *(const v2f*)(freqs + ((size_t)token * 32 + lane16 + 16) * 2);
  const float* vg = qkv + ((size_t)(b * NTOK + token)) * QKVS + 2 * CCH + h * 64;
  r.vv = *(const float4*)(vg + 4 * lane16);
  return r;
}

__device__ __forceinline__ void stage_store(
    const KVregs& r, int tid, float* __restrict__ kdst, float* __restrict__ vdst) {
  const int tok16  = tid >> 4;
  const int lane16 = tid & 15;
  v2f k0, k1;
  k0.x = r.t0.x * r.c0.x - r.t0.y * r.c0.y;
  k0.y = r.t0.y * r.c0.x + r.t0.x * r.c0.y;
  k1.x = r.t1.x * r.c1.x - r.t1.y * r.c1.y;
  k1.y = r.t1.y * r.c1.x + r.t1.x * r.c1.y;
  *(v2f*)(kdst + tok16 * KVP + 2 * lane16)      = k0;
  *(v2f*)(kdst + tok16 * KVP + 2 * lane16 + 32) = k1;
  // V pair-interleave: vdst[(tok/2)*VPP + d*2 + (tok&1)] = V[tok][d]
  float* vp = vdst + (tok16 >> 1) * VPP + (tok16 & 1);
  vp[(4 * lane16 + 0) * 2] = r.vv.x;
  vp[(4 * lane16 + 1) * 2] = r.vv.y;
  vp[(4 * lane16 + 2) * 2] = r.vv.z;
  vp[(4 * lane16 + 3) * 2] = r.vv.w;
}

__global__ __launch_bounds__(256) void attn_wmma_f32(
    const float* __restrict__ qkv, const float* __restrict__ freqs,
    float* __restrict__ out) {
  __shared__ float kbuf[2][16 * KVP];
  __shared__ float vbuf[2][8 * VPP];
  __shared__ float pbuf[8][16 * PP];

  const int tid   = threadIdx.x;
  const int lane  = tid & 31;
  const int wave  = tid >> 5;
  const int laneM = lane & 15;
  const int laneH = lane >> 4;

  const int b  = blockIdx.z;
  const int h  = blockIdx.y;
  const int q0 = (blockIdx.x * 8 + wave) * 16;

  // ---- load Q fragments (A layout) with fused RoPE -----------------------
  const int qtok = q0 + laneM;
  const float* qbase = qkv + ((size_t)(b * NTOK + qtok)) * QKVS + h * 64;
  v2f qa[16];
  #pragma unroll
  for (int s = 0; s < 16; ++s) {
    const int d = 4 * s + 2 * laneH;
    v2f t  = *(const v2f*)(qbase + d);
    v2f cs = *(const v2f*)(freqs + ((size_t)qtok * 32 + (d >> 1)) * 2);
    v2f a;
    a.x = t.x * cs.x - t.y * cs.y;
    a.y = t.y * cs.x + t.x * cs.y;
    qa[s] = a;
  }

  float mrow[8], lrow[8];
  #pragma unroll
  for (int r = 0; r < 8; ++r) { mrow[r] = -1e30f; lrow[r] = 0.f; }
  v8f o0 = {0.f,0.f,0.f,0.f,0.f,0.f,0.f,0.f};
  v8f o1 = o0, o2 = o0, o3 = o0;

  float* pw = &pbuf[wave][0];

  // ---- stage key-block 0 -------------------------------------------------
  {
    KVregs r0 = stage_load(qkv, freqs, b, h, 0, tid);
    stage_store(r0, tid, kbuf[0], vbuf[0]);
  }
  __syncthreads();

  constexpr int NJ = NTOK / 16;  // 128 key blocks
  for (int j = 0; j < NJ; ++j) {
    const int cur = j & 1;

    // issue global loads for the NEXT key block before compute
    KVregs nxt;
    if (j + 1 < NJ) nxt = stage_load(qkv, freqs, b, h, j + 1, tid);

    // ---- S = Q K^T : 16 WMMAs, K fragments from LDS ----------------------
    const float* kb = kbuf[cur];
    v8f sacc = {0.f,0.f,0.f,0.f,0.f,0.f,0.f,0.f};
    #pragma unroll
    for (int s = 0; s < 16; ++s) {
      v2f kfrag = *(const v2f*)(kb + laneM * KVP + 4 * s + 2 * laneH);
      sacc = wmma_f32x4(qa[s], kfrag, sacc);
    }

    // ---- online softmax (row reductions over 16-lane halves) -------------
    #pragma unroll
    for (int r = 0; r < 8; ++r) {
      float sv = sacc[r] * 0.125f;  // D^-0.5
      float mx = sv;
      mx = fmaxf(mx, __shfl_xor(mx, 1));
      mx = fmaxf(mx, __shfl_xor(mx, 2));
      mx = fmaxf(mx, __shfl_xor(mx, 4));
      mx = fmaxf(mx, __shfl_xor(mx, 8));
      const float mnew = fmaxf(mrow[r], mx);
      const float p = __expf(sv - mnew);
      float rs = p;
      rs += __shfl_xor(rs, 1);
      rs += __shfl_xor(rs, 2);
      rs += __shfl_xor(rs, 4);
      rs += __shfl_xor(rs, 8);
      const float alpha = __expf(mrow[r] - mnew);
      mrow[r] = mnew;
      lrow[r] = lrow[r] * alpha + rs;
      o0[r] *= alpha; o1[r] *= alpha; o2[r] *= alpha; o3[r] *= alpha;
      pw[(r + 8 * laneH) * PP + laneM] = p;  // wave-private P tile
    }

    // ---- O += P V : 16 WMMAs (P + pair-interleaved V from LDS) -----------
    const float* vb = vbuf[cur];
    #pragma unroll
    for (int s = 0; s < 4; ++s) {
      v2f pa = *(const v2f*)(pw + laneM * PP + 4 * s + 2 * laneH);
      const float* vp = vb + (2 * s + laneH) * VPP + 2 * laneM;
      v2f vf0 = *(const v2f*)(vp);       o0 = wmma_f32x4(pa, vf0, o0);
      v2f vf1 = *(const v2f*)(vp + 32);  o1 = wmma_f32x4(pa, vf1, o1);
      v2f vf2 = *(const v2f*)(vp + 64);  o2 = wmma_f32x4(pa, vf2, o2);
      v2f vf3 = *(const v2f*)(vp + 96);  o3 = wmma_f32x4(pa, vf3, o3);
    }

    // ---- store staged data for next block, then block barrier ------------
    if (j + 1 < NJ) stage_store(nxt, tid, kbuf[cur ^ 1], vbuf[cur ^ 1]);
    __syncthreads();
  }

  // ---- normalize and store ------------------------------------------------
  #pragma unroll
  for (int r = 0; r < 8; ++r) {
    const float inv = 1.0f / lrow[r];
    const int row = q0 + r + 8 * laneH;
    float* obase = out + ((size_t)(b * NTOK + row)) * CCH + h * 64 + laneM;
    obase[0]  = o0[r] * inv;
    obase[16] = o1[r] * inv;
    obase[32] = o2[r] * inv;
    obase[48] = o3[r] * inv;
  }
}

// ---------------------------------------------------------------------------
// Host launcher
// ---------------------------------------------------------------------------
extern "C" void kernel_launch(void* const* d_in, const int* in_sizes, int n_in,
                              void* d_out, int out_size, void* d_ws, size_t ws_size,
                              hipStream_t stream) {
  (void)in_sizes; (void)n_in; (void)out_size; (void)ws_size;

  const float* x      = (const float*)d_in[0];  // (2,2048,1024)
  const float* freqs  = (const float*)d_in[1];  // (2048,32,2)
  const float* w_qkv  = (const float*)d_in[2];  // (3072,1024)
  const float* w_proj = (const float*)d_in[3];  // (1024,1024)
  const float* b_proj = (const float*)d_in[4];  // (1024,)
  float* out = (float*)d_out;                   // (2,2048,1024)

  float* qkv  = (float*)d_ws;                      // 2*2048*3072 f32 = 50.3 MB
  float* attn = qkv + (size_t)2 * 2048 * 3072;     // 2*2048*1024 f32 = 16.8 MB

  const dim3 blk(256);
  // qkv = x @ w_qkv^T       M=4096, N=3072, K=1024
  gemm16_wmma_f32<<<dim3(3072 / 64, 4096 / 128), blk, 0, stream>>>(
      x, w_qkv, nullptr, qkv, 4096, 3072, 1024);
  // fused RoPE + flash attention (K/V staged in LDS, double-buffered)
  attn_wmma_f32<<<dim3(2048 / 16 / 8, 16, 2), blk, 0, stream>>>(qkv, freqs, attn);
  // out = attn @ w_proj^T + b_proj   M=4096, N=1024, K=1024
  gemm16_wmma_f32<<<dim3(1024 / 64, 4096 / 128), blk, 0, stream>>>(
      attn, w_proj, b_proj, out, 4096, 1024, 1024);
}